// TOMReorientation_17849884082565
// MI455X (gfx1250) — compile-verified
//
#include <hip/hip_runtime.h>
#include <stdint.h>

// Problem geometry (fixed by the reference).
constexpr int Dd   = 128;
constexpr int Hh   = 160;
constexpr int Ww   = 128;
constexpr int HWp  = Hh * Ww;         // 20480
constexpr int NVOX = Dd * HWp;        // 2,621,440  (divisible by 256)
constexpr int BLK  = 256;

// Fast 1-ulp hardware math (v_rcp_f32 / v_rsq_f32 / v_sqrt_f32).
__device__ __forceinline__ float frcp (float x) { return __builtin_amdgcn_rcpf(x);  }
__device__ __forceinline__ float frsq (float x) { return __builtin_amdgcn_rsqf(x);  }
__device__ __forceinline__ float fsqrt(float x) { return __builtin_amdgcn_sqrtf(x); }

// jnp.gradient along one axis: one-sided at the two edges, central*0.5 interior.
// Branch-free: two loads + cndmask'd offsets/scale.
__device__ __forceinline__ float fd_grad(const float* __restrict__ f, int base,
                                         int coord, int last, int stride) {
    const float lo = f[base + ((coord == 0)    ? 0 : -stride)];
    const float hi = f[base + ((coord == last) ? 0 :  stride)];
    const float scale = (coord == 0 || coord == last) ? 1.0f : 0.5f;
    return (hi - lo) * scale;
}

__global__ __launch_bounds__(BLK)
void tom_reorient_kernel(const float* __restrict__ src,
                         const float* __restrict__ flow,
                         float* __restrict__ out)
{
    __shared__ float sv[3 * BLK];

    const int tid = threadIdx.x;
    const int idx = blockIdx.x * BLK + tid;   // grid is exact: NVOX / 256 blocks

    // ---------------------------------------------------------------
    // CDNA5 async DMA: stage this block's src vectors (3 components)
    // into LDS, tracked by ASYNCcnt, fully overlapped with the Jacobi
    // SVD below. Generic LDS pointer low 32 bits == LDS byte offset.
    // ---------------------------------------------------------------
    {
        const uint32_t l0 = (uint32_t)(uintptr_t)(&sv[tid]);
        const uint32_t l1 = l0 + BLK * 4;
        const uint32_t l2 = l0 + 2 * BLK * 4;
        const uint64_t g0 = (uint64_t)(uintptr_t)(src + idx);
        const uint64_t g1 = (uint64_t)(uintptr_t)(src + idx + NVOX);
        const uint64_t g2 = (uint64_t)(uintptr_t)(src + idx + 2 * NVOX);
        asm volatile("global_load_async_to_lds_b32 %0, %1, off"
                     :: "v"(l0), "v"(g0) : "memory");
        asm volatile("global_load_async_to_lds_b32 %0, %1, off"
                     :: "v"(l1), "v"(g1) : "memory");
        asm volatile("global_load_async_to_lds_b32 %0, %1, off"
                     :: "v"(l2), "v"(g2) : "memory");
    }

    // Voxel coordinates (W = 128 is a power of two; H = 160 is not).
    const int w  = idx & (Ww - 1);
    const int dh = idx >> 7;
    const int h  = dh % Hh;
    const int d  = dh / Hh;

    // J = I + grad(flow);  rows = component, cols = direction (d,h,w).
    // B starts as J and gets its columns orthogonalized in-place.
    float B[3][3];
    #pragma unroll
    for (int c = 0; c < 3; ++c) {
        const int base = idx + c * NVOX;
        B[c][0] = fd_grad(flow, base, d, Dd - 1, HWp);
        B[c][1] = fd_grad(flow, base, h, Hh - 1, Ww);
        B[c][2] = fd_grad(flow, base, w, Ww - 1, 1);
        B[c][c] += 1.0f;
    }

    float V[3][3] = {{1.f,0.f,0.f},{0.f,1.f,0.f},{0.f,0.f,1.f}};

    // One-sided Jacobi SVD: orthogonalize columns of B, accumulate V.
    // Cyclic 3x3 Jacobi converges quadratically; 5 sweeps is past fp32
    // round-off. All rotations are branch-free (cndmask) so wave32
    // lanes never diverge; all reciprocals are 1-ulp v_rcp/v_rsq.
    const int PL[3] = {0, 0, 1};
    const int QL[3] = {1, 2, 2};
    #pragma unroll
    for (int sweep = 0; sweep < 5; ++sweep) {
        #pragma unroll
        for (int k = 0; k < 3; ++k) {
            const int p = PL[k], q = QL[k];
            float al = 0.f, be = 0.f, ga = 0.f;
            #pragma unroll
            for (int r = 0; r < 3; ++r) {
                al = fmaf(B[r][p], B[r][p], al);
                be = fmaf(B[r][q], B[r][q], be);
                ga = fmaf(B[r][p], B[r][q], ga);
            }
            const bool rot = (ga != 0.0f);
            // zeta = (be-al)/(2*ga), guarded so ga==0 never makes NaN/inf fma.
            const float zeta = (be - al) * 0.5f * frcp(rot ? ga : 1.0f);
            // t = sign(zeta)/(|zeta| + sqrt(1+zeta^2)); zeta=+-inf -> t=0.
            float t = copysignf(frcp(fabsf(zeta) + fsqrt(fmaf(zeta, zeta, 1.0f))),
                                zeta);
            t = rot ? t : 0.0f;                 // ga==0 -> identity rotation
            const float cs = frsq(fmaf(t, t, 1.0f));
            const float sn = cs * t;
            #pragma unroll
            for (int r = 0; r < 3; ++r) {
                const float bp = B[r][p], bq = B[r][q];
                B[r][p] = fmaf(cs, bp, -sn * bq);
                B[r][q] = fmaf(sn, bp,  cs * bq);
                const float vp = V[r][p], vq = V[r][q];
                V[r][p] = fmaf(cs, vp, -sn * vq);
                V[r][q] = fmaf(sn, vp,  cs * vq);
            }
        }
    }

    // Singular values = column norms of the orthogonalized B; U = B * diag(1/s).
    float s[3], inv[3];
    #pragma unroll
    for (int c = 0; c < 3; ++c) {
        const float nn = fmaf(B[0][c], B[0][c],
                         fmaf(B[1][c], B[1][c], B[2][c] * B[2][c]));
        s[c]   = fsqrt(nn);
        inv[c] = (s[c] > 0.0f) ? frcp(s[c]) : 0.0f;
    }
    // Reference degeneracy test: exact equality of any singular-value pair.
    // (All pairs are compared, so the lack of sorting is irrelevant.)
    const bool degen = (s[0] == s[1]) || (s[0] == s[2]) || (s[1] == s[2]);

    // DMA must have landed before reading v from LDS (lane-private slots,
    // same wave wrote them -> no barrier needed, just the async counter).
    asm volatile("s_wait_asynccnt 0" ::: "memory");
    const float v0 = sv[tid];
    const float v1 = sv[BLK + tid];
    const float v2 = sv[2 * BLK + tid];

    // R = V * U^T  =>  R*v = V * (U^T v),  (U^T v)_j = (B_col_j . v) / s_j.
    float uv[3];
    #pragma unroll
    for (int c = 0; c < 3; ++c) {
        uv[c] = fmaf(B[0][c], v0, fmaf(B[1][c], v1, B[2][c] * v2)) * inv[c];
    }
    const float r0 = fmaf(V[0][0], uv[0], fmaf(V[0][1], uv[1], V[0][2] * uv[2]));
    const float r1 = fmaf(V[1][0], uv[0], fmaf(V[1][1], uv[1], V[1][2] * uv[2]));
    const float r2 = fmaf(V[2][0], uv[0], fmaf(V[2][1], uv[1], V[2][2] * uv[2]));

    out[idx]            = degen ? v0 : r0;
    out[idx + NVOX]     = degen ? v1 : r1;
    out[idx + 2 * NVOX] = degen ? v2 : r2;
}

extern "C" void kernel_launch(void* const* d_in, const int* in_sizes, int n_in,
                              void* d_out, int out_size, void* d_ws, size_t ws_size,
                              hipStream_t stream) {
    (void)in_sizes; (void)n_in; (void)out_size; (void)d_ws; (void)ws_size;
    const float* src  = (const float*)d_in[0];   // [1,3,D,H,W] fp32
    const float* flow = (const float*)d_in[1];   // [1,3,D,H,W] fp32
    float*       out  = (float*)d_out;           // [1,3,D,H,W] fp32

    const dim3 grid(NVOX / BLK);  // 10240 blocks, exact coverage
    const dim3 block(BLK);        // 8 wave32 waves per block
    hipLaunchKernelGGL(tom_reorient_kernel, grid, block, 0, stream,
                       src, flow, out);
}